// SlidingWindowAttention_79388175499574
// MI455X (gfx1250) — compile-verified
//
#include <hip/hip_runtime.h>
#include <hip/hip_bf16.h>

// Sliding-window single-query attention, algebraically refactored:
//   logits[b,h,s] = kv[b,s,:] . qW[b,h,:]   where qW = (scale*q) @ Wk  (per head)
//   out[b,h,:]    = ctx[b,h,:] @ Wv_h^T     where ctx = attn @ kv
// -> 15 GFLOP total instead of 2.2 TFLOP; HBM-bound on streaming `buffer`.
// All GEMMs use fp32 WMMA (V_WMMA_F32_16X16X4_F32), one 16x16 tile per wave32.

typedef __attribute__((ext_vector_type(2))) float v2f;
typedef __attribute__((ext_vector_type(8))) float v8f;

namespace swa {

constexpr int B = 128, W = 512, D = 2048, H = 16, DH = 128;
constexpr float SCALE = 0.08838834764831845f; // Dh^-0.5

__device__ __forceinline__ v8f wmma4(v2f a, v2f b, v8f c) {
  // D = A(16x4 f32) * B(4x16 f32) + C(16x16 f32)
  return __builtin_amdgcn_wmma_f32_16x16x4_f32(false, a, false, b, (short)0, c,
                                               false, false);
}

__device__ __forceinline__ v2f ld2(const float* p) { return *(const v2f*)p; }
__device__ __forceinline__ v2f ld2nt(const float* p) {
  return __builtin_nontemporal_load((const v2f*)p);
}

// ---- K1: q = scale * (x @ Wq^T)   [B,D] ---------------------------------
__global__ void k_qproj(const float* __restrict__ x, const float* __restrict__ wq,
                        float* __restrict__ q) {
  const int wid = (blockIdx.x * blockDim.x + threadIdx.x) >> 5;
  const int lane = threadIdx.x & 31;
  constexpr int NT = D / 16;
  if (wid >= (B / 16) * NT) return;
  const int m0 = (wid / NT) * 16, n0 = (wid % NT) * 16;
  const int rl = lane & 15, ko = (lane >> 4) << 1;
  const float* ap = x + (size_t)(m0 + rl) * D + ko;
  const float* bp = wq + (size_t)(n0 + rl) * D + ko; // B = Wq^T, rows of Wq
  v8f c = {};
#pragma unroll 8
  for (int k = 0; k < D; k += 4) c = wmma4(ld2(ap + k), ld2(bp + k), c);
  const int mb = m0 + ((lane >> 4) << 3), n = n0 + rl;
#pragma unroll
  for (int r = 0; r < 8; ++r) q[(size_t)(mb + r) * D + n] = c[r] * SCALE;
}

// ---- K2: qW[b,h,:] = q[b,h*DH:+DH] @ Wk[h*DH:+DH, :]   [B,H,D] ----------
__global__ void k_qw(const float* __restrict__ q, const float* __restrict__ wk,
                     float* __restrict__ qw) {
  const int wid = (blockIdx.x * blockDim.x + threadIdx.x) >> 5;
  const int lane = threadIdx.x & 31;
  constexpr int NT = D / 16, MT = B / 16;
  if (wid >= H * MT * NT) return;
  int t = wid;
  const int nt = t % NT; t /= NT;
  const int mt = t % MT; const int h = t / MT;
  const int m0 = mt * 16, n0 = nt * 16;
  const int rl = lane & 15, ko = (lane >> 4) << 1;
  const float* ap = q + (size_t)(m0 + rl) * D + h * DH + ko;
  const float* bb = wk + (size_t)(h * DH) * D + n0 + rl; // B[k][n]=Wk[hDH+k][n]
  v8f c = {};
#pragma unroll 4
  for (int k = 0; k < DH; k += 4) {
    const float* p = bb + (size_t)(k + ko) * D;
    v2f bf = {p[0], p[D]};
    c = wmma4(ld2(ap + k), bf, c);
  }
  const int mb = m0 + ((lane >> 4) << 3), n = n0 + rl;
#pragma unroll
  for (int r = 0; r < 8; ++r) qw[((size_t)(mb + r) * H + h) * D + n] = c[r];
}

// ---- K3: logits[b,h,s] = kv[b,s,:] . qW[b,h,:]   stored [B,H,W] ---------
__global__ void k_logits(const float* __restrict__ x, const float* __restrict__ buf,
                         const float* __restrict__ qw, float* __restrict__ lgt) {
  const int wid = (blockIdx.x * blockDim.x + threadIdx.x) >> 5;
  const int lane = threadIdx.x & 31;
  constexpr int MT = W / 16;
  if (wid >= B * MT) return;
  const int b = wid / MT, m0 = (wid % MT) * 16;
  const int rl = lane & 15, ko = (lane >> 4) << 1;
  const int s = m0 + rl; // ring: kv[s] = buffer[s+1] for s<W-1 else x
  const float* ap =
      ((s < W - 1) ? (buf + ((size_t)b * W + s + 1) * D) : (x + (size_t)b * D)) + ko;
  const float* bp = qw + ((size_t)b * H + rl) * D + ko; // N = H = one tile
  v8f c = {};
#pragma unroll 8
  for (int k = 0; k < D; k += 4) c = wmma4(ld2nt(ap + k), ld2(bp + k), c);
  const int mb = m0 + ((lane >> 4) << 3);
  float* op = lgt + ((size_t)b * H + rl) * W; // [b][h=n][s=m]
#pragma unroll
  for (int r = 0; r < 8; ++r) op[mb + r] = c[r];
}

// ---- K4: softmax over s (in place), one block per (b,h) -----------------
__global__ void k_softmax(float* __restrict__ lgt) {
  __shared__ float sm[256];
  float* p = lgt + (size_t)blockIdx.x * swa::W;
  const int t = threadIdx.x;
  const float v0 = p[t], v1 = p[t + 256];
  sm[t] = fmaxf(v0, v1);
  __syncthreads();
  for (int s = 128; s > 0; s >>= 1) {
    if (t < s) sm[t] = fmaxf(sm[t], sm[t + s]);
    __syncthreads();
  }
  const float mx = sm[0];
  __syncthreads();
  const float e0 = __expf(v0 - mx), e1 = __expf(v1 - mx);
  sm[t] = e0 + e1;
  __syncthreads();
  for (int s = 128; s > 0; s >>= 1) {
    if (t < s) sm[t] += sm[t + s];
    __syncthreads();
  }
  const float inv = 1.0f / sm[0];
  p[t] = e0 * inv;
  p[t + 256] = e1 * inv;
}

// ---- K5: ctx[b,h,:] = attn[b,h,:] @ kv[b]   [B,H,D] ---------------------
__global__ void k_ctx(const float* __restrict__ x, const float* __restrict__ buf,
                      const float* __restrict__ attn, float* __restrict__ ctx) {
  const int wid = (blockIdx.x * blockDim.x + threadIdx.x) >> 5;
  const int lane = threadIdx.x & 31;
  constexpr int NT = D / 16;
  if (wid >= B * NT) return;
  const int b = wid / NT, n0 = (wid % NT) * 16;
  const int rl = lane & 15, ko = (lane >> 4) << 1;
  const float* ap = attn + ((size_t)b * H + rl) * W + ko; // M = H = one tile
  const float* xrow = x + (size_t)b * D;
  const float* brow = buf + (size_t)b * W * D;
  const int n = n0 + rl;
  v8f c = {};
#pragma unroll 4
  for (int k = 0; k < W; k += 4) {
    const int s0 = k + ko;
    const float* r0 = (s0 < W - 1) ? brow + (size_t)(s0 + 1) * D : xrow;
    const float* r1 = (s0 + 1 < W - 1) ? brow + (size_t)(s0 + 2) * D : xrow;
    v2f bf = {__builtin_nontemporal_load(r0 + n),
              __builtin_nontemporal_load(r1 + n)};
    c = wmma4(ld2(ap + k), bf, c);
  }
  const int mb = (lane >> 4) << 3;
#pragma unroll
  for (int r = 0; r < 8; ++r) ctx[((size_t)b * H + mb + r) * D + n] = c[r];
}

// ---- K6: out_attn[b, h*DH+d] = ctx[b,h,:] . Wv[h*DH+d, :]   [B,D] -------
__global__ void k_vproj(const float* __restrict__ ctx, const float* __restrict__ wv,
                        float* __restrict__ oat) {
  const int wid = (blockIdx.x * blockDim.x + threadIdx.x) >> 5;
  const int lane = threadIdx.x & 31;
  constexpr int NT = DH / 16, MT = B / 16;
  if (wid >= H * MT * NT) return;
  int t = wid;
  const int nt = t % NT; t /= NT;
  const int mt = t % MT; const int h = t / MT;
  const int m0 = mt * 16, n0 = nt * 16;
  const int rl = lane & 15, ko = (lane >> 4) << 1;
  const float* ap = ctx + ((size_t)(m0 + rl) * H + h) * D + ko; // row stride H*D
  const float* bp = wv + (size_t)(h * DH + n0 + rl) * D + ko;   // B = Wv_h^T
  v8f c = {};
#pragma unroll 8
  for (int k = 0; k < D; k += 4) c = wmma4(ld2(ap + k), ld2(bp + k), c);
  const int mb = m0 + ((lane >> 4) << 3), n = h * DH + n0 + rl;
#pragma unroll
  for (int r = 0; r < 8; ++r) oat[(size_t)(mb + r) * D + n] = c[r];
}

// ---- K7: out2 = out_attn @ Wo^T   [B,D] ---------------------------------
__global__ void k_oproj(const float* __restrict__ oat, const float* __restrict__ wo,
                        float* __restrict__ out2) {
  const int wid = (blockIdx.x * blockDim.x + threadIdx.x) >> 5;
  const int lane = threadIdx.x & 31;
  constexpr int NT = D / 16;
  if (wid >= (B / 16) * NT) return;
  const int m0 = (wid / NT) * 16, n0 = (wid % NT) * 16;
  const int rl = lane & 15, ko = (lane >> 4) << 1;
  const float* ap = oat + (size_t)(m0 + rl) * D + ko;
  const float* bp = wo + (size_t)(n0 + rl) * D + ko;
  v8f c = {};
#pragma unroll 8
  for (int k = 0; k < D; k += 4) c = wmma4(ld2(ap + k), ld2(bp + k), c);
  const int mb = m0 + ((lane >> 4) << 3), n = n0 + rl;
#pragma unroll
  for (int r = 0; r < 8; ++r) out2[(size_t)(mb + r) * D + n] = c[r];
}

// ---- K8: y = x + sigmoid([x|out2] @ Wg^T + bg) * out2 -------------------
__global__ void k_gate(const float* __restrict__ x, const float* __restrict__ out2,
                       const float* __restrict__ wg, const float* __restrict__ bg,
                       float* __restrict__ y) {
  const int wid = (blockIdx.x * blockDim.x + threadIdx.x) >> 5;
  const int lane = threadIdx.x & 31;
  constexpr int NT = D / 16;
  if (wid >= (B / 16) * NT) return;
  const int m0 = (wid / NT) * 16, n0 = (wid % NT) * 16;
  const int rl = lane & 15, ko = (lane >> 4) << 1;
  const float* ax = x + (size_t)(m0 + rl) * D + ko;
  const float* ao = out2 + (size_t)(m0 + rl) * D + ko;
  const float* bp = wg + (size_t)(n0 + rl) * (2 * D) + ko; // Wg rows length 2D
  v8f c = {};
#pragma unroll 8
  for (int k = 0; k < D; k += 4) c = wmma4(ld2(ax + k), ld2(bp + k), c);
#pragma unroll 8
  for (int k = 0; k < D; k += 4) c = wmma4(ld2(ao + k), ld2(bp + D + k), c);
  const int mb = m0 + ((lane >> 4) << 3), n = n0 + rl;
#pragma unroll
  for (int r = 0; r < 8; ++r) {
    const size_t idx = (size_t)(mb + r) * D + n;
    const float z = c[r] + bg[n];
    const float g = 1.0f / (1.0f + __expf(-z));
    y[idx] = x[idx] + g * out2[idx];
  }
}

} // namespace swa

extern "C" void kernel_launch(void* const* d_in, const int* in_sizes, int n_in,
                              void* d_out, int out_size, void* d_ws, size_t ws_size,
                              hipStream_t stream) {
  using namespace swa;
  const float* x   = (const float*)d_in[0];
  const float* buf = (const float*)d_in[1];
  const float* wq  = (const float*)d_in[2];
  const float* wk  = (const float*)d_in[3];
  const float* wv  = (const float*)d_in[4];
  const float* wo  = (const float*)d_in[5];
  const float* wg  = (const float*)d_in[6];
  const float* bg  = (const float*)d_in[7];
  float* out = (float*)d_out;

  float* ws   = (float*)d_ws;
  float* q    = ws;                       // B*D
  float* qw   = q + (size_t)B * D;        // B*H*D
  float* lgt  = qw + (size_t)B * H * D;   // B*H*W (softmax in place)
  float* ctx  = lgt + (size_t)B * H * W;  // B*H*D
  float* oat  = ctx + (size_t)B * H * D;  // B*D
  float* out2 = oat + (size_t)B * D;      // B*D  (total ~41 MB)

  const int TPB = 128; // 4 waves/block, 1 WMMA tile per wave
  k_qproj  <<<(B / 16) * (D / 16) / 4, TPB, 0, stream>>>(x, wq, q);
  k_qw     <<<H * (B / 16) * (D / 16) / 4, TPB, 0, stream>>>(q, wk, qw);
  k_logits <<<B * (W / 16) / 4, TPB, 0, stream>>>(x, buf, qw, lgt);
  k_softmax<<<B * H, 256, 0, stream>>>(lgt);
  k_ctx    <<<B * (D / 16) / 4, TPB, 0, stream>>>(x, buf, lgt, ctx);
  k_vproj  <<<H * (B / 16) * (DH / 16) / 4, TPB, 0, stream>>>(ctx, wv, oat);
  k_oproj  <<<(B / 16) * (D / 16) / 4, TPB, 0, stream>>>(oat, wo, out2);
  k_gate   <<<(B / 16) * (D / 16) / 4, TPB, 0, stream>>>(x, out2, wg, bg, out);
}